// ObjectDetectionPredictionDecoder_66099546685869
// MI455X (gfx1250) — compile-verified
//
#include <hip/hip_runtime.h>
#include <stdint.h>

// ---------------------------------------------------------------------------
// ObjectDetectionPredictionDecoder for MI455X (gfx1250)
// Phase 1: decode kernel  - async global->LDS staging of 84-f32 anchor rows,
//                           max/argmax over 80 logits, box decode.
// Phase 2: NMS kernel     - exact greedy NMS via histogram-threshold candidate
//                           extraction + bitonic sort + wave32 greedy.
// ---------------------------------------------------------------------------

#define STRIDE     84      // 4 box + 80 classes
#define NCLS       80
#define DEC_TILE   128     // anchors per decode block
#define DEC_THREADS 128
#define NMS_THREADS 1024
#define NB         1024    // score histogram bins
#define CAP        2048    // candidate buffer (power of two, bitonic size)
#define MAXDET     100
#define CONF_TH    0.05f
#define IOU_TH     0.5f

// ---------------------------------------------------------------------------
// Decode kernel: grid = (ceil(A/DEC_TILE), B), block = DEC_THREADS
// ---------------------------------------------------------------------------
__global__ __launch_bounds__(DEC_THREADS)
void decode_kernel(const float* __restrict__ pred,
                   const float* __restrict__ anchors,
                   float* __restrict__ boxes6,
                   float* __restrict__ scores,
                   int A)
{
    __shared__ float tile[DEC_TILE * STRIDE];   // 128*84*4 = 43008 B

    const int tid       = threadIdx.x;
    const int b         = blockIdx.y;
    const int tileStart = blockIdx.x * DEC_TILE;
    const int tileAnch  = min(DEC_TILE, A - tileStart);
    if (tileAnch <= 0) return;

    // --- async copy tile of predictions into LDS (16B chunks) ------------
    // row of one anchor = 84 f32 = 336 B = 21 x 16B chunks, 16B-aligned.
    const int chunks = tileAnch * (STRIDE / 4);
    const float* gbase = pred + ((size_t)b * A + tileStart) * STRIDE;
    const uint32_t ldsBase = (uint32_t)(size_t)(&tile[0]); // low 32 bits = LDS byte addr

    for (int c = tid; c < chunks; c += DEC_THREADS) {
        unsigned long long gaddr = (unsigned long long)(size_t)gbase
                                 + (unsigned long long)c * 16ull;
        uint32_t laddr = ldsBase + (uint32_t)c * 16u;
        asm volatile("global_load_async_to_lds_b128 %0, %1, off"
                     :: "v"(laddr), "v"(gaddr) : "memory");
    }
    asm volatile("s_wait_asynccnt 0" ::: "memory");
    __syncthreads();

    // --- per-thread anchor processing from LDS ---------------------------
    const int a = tileStart + tid;
    if (tid < tileAnch && a < A) {
        const float* row = &tile[tid * STRIDE];
        const float4 bp  = *(const float4*)row;           // box preds t_x,t_y,t_w,t_h

        // max/argmax over 80 logits (sigmoid monotonic -> same argmax;
        // strict '>' in ascending order == first-max like jnp.argmax)
        const float4* cr = (const float4*)(row + 4);
        float m = -3.4e38f; int mi = 0;
        #pragma unroll
        for (int j4 = 0; j4 < NCLS / 4; ++j4) {
            float4 v = cr[j4];
            int base = j4 * 4;
            if (v.x > m) { m = v.x; mi = base + 0; }
            if (v.y > m) { m = v.y; mi = base + 1; }
            if (v.z > m) { m = v.z; mi = base + 2; }
            if (v.w > m) { m = v.w; mi = base + 3; }
        }
        float conf = 1.0f / (1.0f + expf(-m));

        // decode: b = t * [.1,.1,.2,.2]; xy = b.xy*size + center; wh = exp(b.zw)*size
        const float4 av = ((const float4*)anchors)[a];    // cx,cy,w,h of anchor
        float cx = (bp.x * 0.1f) * av.z + av.x;
        float cy = (bp.y * 0.1f) * av.w + av.y;
        float w  = expf(bp.z * 0.2f) * av.z;
        float h  = expf(bp.w * 0.2f) * av.w;

        size_t o  = (size_t)b * A + a;
        float* r6 = boxes6 + o * 6;
        r6[0] = cx; r6[1] = cy; r6[2] = w; r6[3] = h;
        r6[4] = (float)mi; r6[5] = conf;
        scores[o] = (conf >= CONF_TH) ? conf : 0.0f;
    }
}

// ---------------------------------------------------------------------------
// NMS kernel: grid = B blocks, block = NMS_THREADS (32 waves of wave32)
// Exact greedy NMS == iterated-argmax NMS of the reference.
// ---------------------------------------------------------------------------
__global__ __launch_bounds__(NMS_THREADS)
void nms_kernel(const float* __restrict__ boxes6,
                float* __restrict__ scores,   // mutated (zero processed)
                float* __restrict__ out,      // B x 100 x 6
                int A)
{
    __shared__ unsigned int       bins[NB];
    __shared__ unsigned long long cand[CAP];
    __shared__ float selX1[MAXDET], selY1[MAXDET], selX2[MAXDET],
                     selY2[MAXDET], selAr[MAXDET], selCl[MAXDET];
    __shared__ unsigned int candCount, remaining, thrBin, selCount;

    const int tid = threadIdx.x;
    const int b   = blockIdx.x;
    float* sc         = scores + (size_t)b * A;
    const float* bx6  = boxes6 + (size_t)b * A * 6;
    float* outp       = out    + (size_t)b * MAXDET * 6;

    if (tid == 0) selCount = 0;
    __syncthreads();

    for (int round = 0; round < 256; ++round) {
        // ---- reset --------------------------------------------------------
        for (int i = tid; i < NB; i += NMS_THREADS) bins[i] = 0;
        if (tid == 0) { candCount = 0; remaining = 0; }
        __syncthreads();

        // ---- pass 1: score histogram -------------------------------------
        unsigned int localN = 0;
        for (int i = tid; i < A; i += NMS_THREADS) {
            float s = sc[i];
            if (s > 0.0f) {
                int bin = (int)(s * (float)NB);
                bin = bin > (NB - 1) ? (NB - 1) : bin;
                atomicAdd(&bins[bin], 1u);
                ++localN;
            }
        }
        if (localN) atomicAdd(&remaining, localN);
        __syncthreads();

        // ---- pick threshold bin: largest suffix with count <= CAP --------
        if (tid == 0) {
            unsigned int cum = 0; int thr = 0;
            for (int bb = NB - 1; bb >= 0; --bb) {
                unsigned int c = bins[bb];
                if (cum + c > CAP) {
                    thr = (cum > 0) ? (bb + 1) : bb;  // fat single bin: take capped
                    cum = CAP;
                    break;
                }
                cum += c;
            }
            thrBin = (unsigned int)thr;
        }
        __syncthreads();
        if (remaining == 0) break;                       // reference: rest are idx 0
        const int thr = (int)thrBin;

        // ---- pass 2: collect candidates >= threshold ---------------------
        for (int i = tid; i < A; i += NMS_THREADS) {
            float s = sc[i];
            if (s > 0.0f) {
                int bin = (int)(s * (float)NB);
                bin = bin > (NB - 1) ? (NB - 1) : bin;
                if (bin >= thr) {
                    unsigned int pos = atomicAdd(&candCount, 1u);
                    if (pos < CAP) {
                        // key: score desc, then index asc (via complement)
                        cand[pos] = ((unsigned long long)__float_as_uint(s) << 32)
                                  | (unsigned long long)(0xFFFFFFFFu - (unsigned int)i);
                        sc[i] = 0.0f;                    // processed this round
                    }
                }
            }
        }
        __syncthreads();
        int n = (int)candCount; if (n > CAP) n = CAP;
        for (int i = n + tid; i < CAP; i += NMS_THREADS) cand[i] = 0ull; // pad
        __syncthreads();

        // ---- bitonic sort CAP keys, descending (deterministic order) -----
        for (int k = 2; k <= CAP; k <<= 1) {
            for (int j = k >> 1; j > 0; j >>= 1) {
                for (int i = tid; i < CAP; i += NMS_THREADS) {
                    int ixj = i ^ j;
                    if (ixj > i) {
                        unsigned long long va = cand[i], vb = cand[ixj];
                        bool up = ((i & k) == 0);
                        if (up ? (va < vb) : (va > vb)) { cand[i] = vb; cand[ixj] = va; }
                    }
                }
                __syncthreads();
            }
        }

        // ---- greedy selection on wave 0 ----------------------------------
        if (tid < 32) {
            const int lane = tid;
            volatile float* vX1 = selX1; volatile float* vY1 = selY1;
            volatile float* vX2 = selX2; volatile float* vY2 = selY2;
            volatile float* vAr = selAr; volatile float* vCl = selCl;
            int sel = (int)selCount;
            for (int c = 0; c < n && sel < MAXDET; ++c) {
                unsigned long long key = cand[c];
                float s = __uint_as_float((unsigned int)(key >> 32));
                if (s <= 0.0f) break;
                unsigned int idx = 0xFFFFFFFFu - (unsigned int)(key & 0xFFFFFFFFull);
                const float* r6 = bx6 + (size_t)idx * 6;
                float cx = r6[0], cy = r6[1], w = r6[2], h = r6[3], cls = r6[4];
                float x1 = cx - 0.5f * w, y1 = cy - 0.5f * h;
                float x2 = cx + 0.5f * w, y2 = cy + 0.5f * h;
                float area = w * h;
                int hit = 0;
                for (int k2 = lane; k2 < sel; k2 += 32) {
                    if (vCl[k2] == cls) {
                        float lx = fmaxf(x1, vX1[k2]), ly = fmaxf(y1, vY1[k2]);
                        float rx = fminf(x2, vX2[k2]), ry = fminf(y2, vY2[k2]);
                        float iw = fmaxf(rx - lx, 0.0f), ih = fmaxf(ry - ly, 0.0f);
                        float inter = iw * ih;
                        float iou = inter / (area + vAr[k2] - inter + 1e-8f);
                        if (iou > IOU_TH) hit = 1;
                    }
                }
                unsigned int mask = __builtin_amdgcn_ballot_w32(hit != 0);
                if (mask == 0u) {
                    if (lane == 0) {
                        vX1[sel] = x1; vY1[sel] = y1; vX2[sel] = x2; vY2[sel] = y2;
                        vAr[sel] = area; vCl[sel] = cls;
                        float* orow = outp + (size_t)sel * 6;
                        orow[0] = cx; orow[1] = cy; orow[2] = w; orow[3] = h;
                        orow[4] = cls; orow[5] = s;
                    }
                    ++sel;   // uniform across wave
                }
            }
            if (lane == 0) selCount = (unsigned int)sel;
        }
        __syncthreads();
        if (selCount >= MAXDET) break;
    }

    // ---- fill remaining rows: argmax over all-zero scores -> idx 0, conf 0
    __syncthreads();
    const unsigned int selF = selCount;
    for (unsigned int r = selF + tid; r < MAXDET; r += NMS_THREADS) {
        float* orow = outp + (size_t)r * 6;
        orow[0] = bx6[0]; orow[1] = bx6[1]; orow[2] = bx6[2];
        orow[3] = bx6[3]; orow[4] = bx6[4]; orow[5] = 0.0f;
    }
}

// ---------------------------------------------------------------------------
extern "C" void kernel_launch(void* const* d_in, const int* in_sizes, int n_in,
                              void* d_out, int out_size, void* d_ws, size_t ws_size,
                              hipStream_t stream)
{
    const float* pred    = (const float*)d_in[0];   // (B, A, 84) f32
    const float* anchors = (const float*)d_in[1];   // (A, 4)     f32
    float* out = (float*)d_out;                     // (B, 100, 6)

    const int A = in_sizes[1] / 4;
    const int B = in_sizes[0] / (A * STRIDE);

    float* boxes6 = (float*)d_ws;                   // B*A*6 f32
    float* scores = boxes6 + (size_t)B * A * 6;     // B*A   f32  (~17 MB total)

    dim3 g1((A + DEC_TILE - 1) / DEC_TILE, B);
    decode_kernel<<<g1, DEC_THREADS, 0, stream>>>(pred, anchors, boxes6, scores, A);
    nms_kernel<<<B, NMS_THREADS, 0, stream>>>(boxes6, scores, out, A);
}